// GraphSAGE_13778255085862
// MI455X (gfx1250) — compile-verified
//
#include <hip/hip_runtime.h>
#include <math.h>

#define NU 100000
#define NI 50000
#define NE 1000000
#define HD 128
#define LDSP 132   // padded pitch for the 16x128 h-tile staged in LDS

typedef __attribute__((ext_vector_type(2))) float v2f;
typedef __attribute__((ext_vector_type(8))) float v8f;

// D = A(16x4 f32) * B(4x16 f32) + C(16x16 f32), wave32 WMMA
__device__ __forceinline__ v8f wmma_f32(v2f a, v2f b, v8f c) {
    return __builtin_amdgcn_wmma_f32_16x16x4_f32(
        /*neg_a=*/false, a, /*neg_b=*/false, b,
        /*c_mod=*/(short)0, c, /*reuse_a=*/false, /*reuse_b=*/false);
}

// ---------------------------------------------------------------- zero scratch
__global__ void k_zero(float* __restrict__ p, int n) {
    int i = blockIdx.x * blockDim.x + threadIdx.x;
    int stride = gridDim.x * blockDim.x;
    for (; i < n; i += stride) p[i] = 0.0f;
}

// ---------------------------------------------------------------- degrees
__global__ void k_degree(const int* __restrict__ src, const int* __restrict__ dst,
                         float* __restrict__ deg_u, float* __restrict__ deg_i) {
    int e = blockIdx.x * blockDim.x + threadIdx.x;
    if (e >= NE) return;
    atomicAdd(&deg_u[src[e]], 1.0f);
    atomicAdd(&deg_i[dst[e]], 1.0f);
}

__global__ void k_invsqrt(float* __restrict__ d, int n) {
    int i = blockIdx.x * blockDim.x + threadIdx.x;
    if (i < n) d[i] = rsqrtf(fmaxf(d[i], 1.0f));
}

// ---------------------------------------------------------------- edge scatter
// one wave per edge; 32 lanes x float4 = 128 floats
__global__ void __launch_bounds__(256)
k_scatter(const int* __restrict__ src, const int* __restrict__ dst,
          const float* __restrict__ xu, const float* __restrict__ xi,
          const float* __restrict__ inv_du, const float* __restrict__ inv_di,
          float* __restrict__ agg_u, float* __restrict__ agg_i) {
    int lane = threadIdx.x & 31;
    int wave = threadIdx.x >> 5;
    int e = blockIdx.x * 8 + wave;
    if (e >= NE) return;
    int s = src[e], d = dst[e];
    float su = inv_du[s];
    float si = inv_di[d];
    const float4 mu = ((const float4*)(xu + (size_t)s * HD))[lane];
    const float4 mi = ((const float4*)(xi + (size_t)d * HD))[lane];
    float* ai = agg_i + (size_t)d * HD + lane * 4;
    float* au = agg_u + (size_t)s * HD + lane * 4;
    atomicAdd(ai + 0, mu.x * su); atomicAdd(ai + 1, mu.y * su);
    atomicAdd(ai + 2, mu.z * su); atomicAdd(ai + 3, mu.w * su);
    atomicAdd(au + 0, mi.x * si); atomicAdd(au + 1, mi.y * si);
    atomicAdd(au + 2, mi.z * si); atomicAdd(au + 3, mi.w * si);
}

// ---------------------------------------------------------------- fused node update
// h = relu((agg * inv) @ Wa + ba)     (K = 128)
// t = [x, h] @ Wc + bc                (K = 256, h staged in LDS)
// block = 8 waves; block owns 16 rows; wave w owns output cols [16w, 16w+16)
__global__ void __launch_bounds__(256)
k_node_update(const float* __restrict__ agg, const float* __restrict__ inv,
              const float* __restrict__ x,
              const float* __restrict__ Wa, const float* __restrict__ ba,
              const float* __restrict__ Wc, const float* __restrict__ bc,
              float* __restrict__ tout) {
    __shared__ float sh[16 * LDSP];

    const int lane = threadIdx.x & 31;
    const int wave = threadIdx.x >> 5;
    const int m0   = blockIdx.x * 16;
    const int n0   = wave * 16;
    const int row  = lane & 15;          // A-fragment: M index
    const int koff = (lane >> 4) * 2;    // A/B-fragment: K sub-offset per half-wave
    const int col  = lane & 15;          // B/C/D-fragment: N index
    const int rbase = (lane >> 4) * 8;   // C/D-fragment: M base per half-wave

    const size_t arow = (size_t)(m0 + row) * HD;
    const float  sc   = inv[m0 + row];

    // ---- GEMM1: h-tile = (agg * inv) @ Wa
    v8f acc = {};
#pragma unroll 8
    for (int k0 = 0; k0 < HD; k0 += 4) {
        const int k = k0 + koff;
        v2f a; a.x = agg[arow + k] * sc;            a.y = agg[arow + k + 1] * sc;
        v2f b; b.x = Wa[(size_t)k * HD + n0 + col]; b.y = Wa[(size_t)(k + 1) * HD + n0 + col];
        acc = wmma_f32(a, b, acc);
    }
    // bias + relu, stage h-tile to LDS (full 16x128 per block)
    const float bias_a = ba[n0 + col];
#pragma unroll
    for (int r = 0; r < 8; ++r) {
        float h = acc[r] + bias_a;
        h = h > 0.0f ? h : 0.0f;
        sh[(rbase + r) * LDSP + n0 + col] = h;
    }
    __syncthreads();

    // ---- GEMM2: t-tile = x @ Wc[0:128] + h @ Wc[128:256]
    v8f acc2 = {};
#pragma unroll 8
    for (int k0 = 0; k0 < HD; k0 += 4) {            // x part from global
        const int k = k0 + koff;
        v2f a; a.x = x[arow + k];                   a.y = x[arow + k + 1];
        v2f b; b.x = Wc[(size_t)k * HD + n0 + col]; b.y = Wc[(size_t)(k + 1) * HD + n0 + col];
        acc2 = wmma_f32(a, b, acc2);
    }
#pragma unroll 8
    for (int k0 = 0; k0 < HD; k0 += 4) {            // h part from LDS
        const int k = k0 + koff;
        v2f a; a.x = sh[row * LDSP + k];            a.y = sh[row * LDSP + k + 1];
        v2f b; b.x = Wc[(size_t)(HD + k) * HD + n0 + col];
               b.y = Wc[(size_t)(HD + k + 1) * HD + n0 + col];
        acc2 = wmma_f32(a, b, acc2);
    }
    const float bias_c = bc[n0 + col];
#pragma unroll
    for (int r = 0; r < 8; ++r)
        tout[(size_t)(m0 + rbase + r) * HD + n0 + col] = acc2[r] + bias_c;
}

// ---------------------------------------------------------------- edge predictor
// one wave per edge: score = softmax([t_u[src], t_i[dst]] @ Wp + bp)
__global__ void __launch_bounds__(256)
k_predict(const int* __restrict__ src, const int* __restrict__ dst,
          const float* __restrict__ tu, const float* __restrict__ ti,
          const float* __restrict__ Wp, const float* __restrict__ bp,
          float* __restrict__ out) {
    int lane = threadIdx.x & 31;
    int wave = threadIdx.x >> 5;
    int e = blockIdx.x * 8 + wave;
    if (e >= NE) return;
    int s = src[e], d = dst[e];
    const float4 u = ((const float4*)(tu + (size_t)s * HD))[lane];
    const float4 v = ((const float4*)(ti + (size_t)d * HD))[lane];
    const float2* W = (const float2*)Wp;   // row k -> (Wp[k][0], Wp[k][1])
    int k = lane * 4;
    float p0 = 0.0f, p1 = 0.0f;
    float2 w;
    w = W[k + 0];      p0 += u.x * w.x; p1 += u.x * w.y;
    w = W[k + 1];      p0 += u.y * w.x; p1 += u.y * w.y;
    w = W[k + 2];      p0 += u.z * w.x; p1 += u.z * w.y;
    w = W[k + 3];      p0 += u.w * w.x; p1 += u.w * w.y;
    w = W[HD + k + 0]; p0 += v.x * w.x; p1 += v.x * w.y;
    w = W[HD + k + 1]; p0 += v.y * w.x; p1 += v.y * w.y;
    w = W[HD + k + 2]; p0 += v.z * w.x; p1 += v.z * w.y;
    w = W[HD + k + 3]; p0 += v.w * w.x; p1 += v.w * w.y;
#pragma unroll
    for (int off = 16; off > 0; off >>= 1) {
        p0 += __shfl_xor(p0, off, 32);
        p1 += __shfl_xor(p1, off, 32);
    }
    if (lane == 0) {
        float s0 = p0 + bp[0], s1 = p1 + bp[1];
        float m = fmaxf(s0, s1);
        float e0 = __expf(s0 - m), e1 = __expf(s1 - m);
        float inv = 1.0f / (e0 + e1);
        out[(size_t)e * 2 + 0] = e0 * inv;
        out[(size_t)e * 2 + 1] = e1 * inv;
    }
}

// ---------------------------------------------------------------- launch
extern "C" void kernel_launch(void* const* d_in, const int* in_sizes, int n_in,
                              void* d_out, int out_size, void* d_ws, size_t ws_size,
                              hipStream_t stream) {
    const float* x_user = (const float*)d_in[0];
    const float* x_item = (const float*)d_in[1];
    const float* W0 = (const float*)d_in[2];
    const float* b0 = (const float*)d_in[3];
    const float* W1 = (const float*)d_in[4];
    const float* b1 = (const float*)d_in[5];
    const float* W2 = (const float*)d_in[6];
    const float* b2 = (const float*)d_in[7];
    const float* W3 = (const float*)d_in[8];
    const float* b3 = (const float*)d_in[9];
    const float* Wp = (const float*)d_in[10];
    const float* bp = (const float*)d_in[11];
    const int* esrc = (const int*)d_in[12];
    const int* edst = (const int*)d_in[13];
    float* out = (float*)d_out;

    // workspace layout (all 16B-aligned offsets)
    float* ws    = (float*)d_ws;
    float* deg_u = ws;                         // NU   (becomes inv_du)
    float* deg_i = deg_u + NU;                 // NI   (becomes inv_di)
    float* agg_u = deg_i + NI;                 // NU*HD
    float* agg_i = agg_u + (size_t)NU * HD;    // NI*HD
    float* t_u   = agg_i + (size_t)NI * HD;    // NU*HD
    float* t_i   = t_u   + (size_t)NU * HD;    // NI*HD

    const int nzero = NU + NI + (NU + NI) * HD;
    k_zero<<<2048, 256, 0, stream>>>(ws, nzero);
    k_degree<<<(NE + 255) / 256, 256, 0, stream>>>(esrc, edst, deg_u, deg_i);
    k_invsqrt<<<(NU + NI + 255) / 256, 256, 0, stream>>>(ws, NU + NI);
    k_scatter<<<NE / 8, 256, 0, stream>>>(esrc, edst, x_user, x_item,
                                          deg_u, deg_i, agg_u, agg_i);
    // items: h_item from W0/b0, t_item from W3/b3
    k_node_update<<<NI / 16, 256, 0, stream>>>(agg_i, deg_i, x_item,
                                               W0, b0, W3, b3, t_i);
    // users: h_user from W1/b1, t_user from W2/b2
    k_node_update<<<NU / 16, 256, 0, stream>>>(agg_u, deg_u, x_user,
                                               W1, b1, W2, b2, t_u);
    k_predict<<<NE / 8, 256, 0, stream>>>(esrc, edst, t_u, t_i, Wp, bp, out);
}